// DualPathMambaBottleneck_58342835749266
// MI455X (gfx1250) — compile-verified
//
#include <hip/hip_runtime.h>

#define BQ   2
#define DIMQ 256
#define LQ   1024
#define HALFQ 128
#define DIQ  256
#define DSQ  16
#define DTRQ 8
#define K1Q  4

typedef __attribute__((ext_vector_type(2))) float v2f;
typedef __attribute__((ext_vector_type(8))) float v8f;

__device__ __forceinline__ float silu_f(float x)     { return x / (1.0f + expf(-x)); }
__device__ __forceinline__ float softplus_f(float x) { return (x > 20.0f) ? x : log1pf(expf(x)); }

// ---------------------------------------------------------------------------
// Strided/batched fp32 GEMM on WMMA 16x16x4_f32, compile-time strides.
// D[bb][m][n] = act( sum_k A[bb][m][k]*B[bidx][k][n] + bias[bidx][n] ) (+=D)
// One wave computes one 16x16 tile; all addressing is pointer-increment.
// Partial N tiles (NLIM % 16 != 0) handled branch-free via address clamp +
// multiplicative 0/1 mask so EXEC stays all-ones around the WMMA.
// ---------------------------------------------------------------------------
template<int SAM, int SAK, int SBK, int SBN, int KK, int SDM, int NLIM, int ACCUM, int ACT>
__global__ void k_gemm(const float* __restrict__ A, long batchA,
                       const float* __restrict__ Bm, long batchB, int bmodB,
                       const float* __restrict__ bias, long batchBias,
                       float* __restrict__ D, long batchD,
                       int nbatch, int Mt, int Nt)
{
    constexpr bool PARTIAL = (NLIM % 16) != 0;

    int wave = (int)(((long)blockIdx.x * blockDim.x + threadIdx.x) >> 5);
    int total = nbatch * Mt * Nt;
    if (wave >= total) return;
    int nt = wave % Nt;
    int mt = (wave / Nt) % Mt;
    int bb = wave / (Nt * Mt);

    int lane = threadIdx.x & 31;
    int hl   = lane >> 4;     // half-wave -> which K pair
    int idx  = lane & 15;     // m for A loads, n for B loads / D column

    int bidx = bmodB ? (bb % bmodB) : bb;
    int ncol = nt * 16 + idx;

    int   ncl = ncol;
    float msk = 1.0f;
    if (PARTIAL && ncol >= NLIM) { ncl = NLIM - 1; msk = 0.0f; }

    const float* Ap = A + (long)bb * batchA + (long)(mt * 16 + idx) * SAM + (long)(2 * hl) * SAK;
    const float* Bp = Bm + (long)bidx * batchB + (long)ncl * SBN + (long)(2 * hl) * SBK;

    v8f c = {0.f,0.f,0.f,0.f,0.f,0.f,0.f,0.f};
    #pragma unroll 4
    for (int k0 = 0; k0 < KK; k0 += 4) {
        v2f a, b;
        a.x = Ap[0];
        a.y = Ap[SAK];
        b.x = Bp[0];
        b.y = Bp[SBK];
        if (PARTIAL) { b.x *= msk; b.y *= msk; }
        c = __builtin_amdgcn_wmma_f32_16x16x4_f32(false, a, false, b, (short)0, c, false, false);
        Ap += 4 * SAK;
        Bp += 4 * SBK;
    }

    if (PARTIAL && ncol >= NLIM) return;
    float bv = bias ? bias[(long)bidx * batchBias + ncol] : 0.f;
    float* Dp = D + (long)bb * batchD + ncol;
    #pragma unroll
    for (int r = 0; r < 8; ++r) {
        int row = mt * 16 + hl * 8 + r;
        float v = c[r] + bv;
        if (ACT == 1) v = softplus_f(v);
        long off = (long)row * SDM;
        if (ACCUM) Dp[off] += v; else Dp[off] = v;
    }
}

// ---------------------------------------------------------------------------
// LayerNorm over last dim C (blockDim == C), one row per block.
// ---------------------------------------------------------------------------
__global__ void k_ln(const float* __restrict__ in, const float* __restrict__ w,
                     const float* __restrict__ bias, float* __restrict__ out, int C)
{
    extern __shared__ float sm[];
    long row = blockIdx.x;
    int  t   = threadIdx.x;
    float v  = in[row * C + t];
    sm[t] = v; __syncthreads();
    for (int s = C >> 1; s > 0; s >>= 1) { if (t < s) sm[t] += sm[t + s]; __syncthreads(); }
    float mean = sm[0] / C; __syncthreads();
    float d = v - mean;
    sm[t] = d * d; __syncthreads();
    for (int s = C >> 1; s > 0; s >>= 1) { if (t < s) sm[t] += sm[t + s]; __syncthreads(); }
    float var = sm[0] / C;
    out[row * C + t] = d * rsqrtf(var + 1e-5f) * w[t] + bias[t];
}

// ---------------------------------------------------------------------------
// Mamba causal conv1d (K1=4) over u half of uz, + bias, SiLU.
// ---------------------------------------------------------------------------
__global__ void k_conv1d_silu(const float* __restrict__ uz, const float* __restrict__ cw,
                              const float* __restrict__ cb, float* __restrict__ u)
{
    int gid = blockIdx.x * blockDim.x + threadIdx.x;
    if (gid >= BQ * LQ * DIQ) return;
    int d = gid % DIQ;
    int l = (gid / DIQ) % LQ;
    int b = gid / (DIQ * LQ);
    float acc = cb[d];
    #pragma unroll
    for (int k = 0; k < K1Q; ++k) {
        int ls = l + k - (K1Q - 1);
        if (ls >= 0) acc += uz[((long)(b * LQ + ls)) * 512 + d] * cw[d * K1Q + k];
    }
    u[gid] = silu_f(acc);
}

// ---------------------------------------------------------------------------
// VSS depthwise 3x3 conv (pad 1) + bias + SiLU.  uz channel-last input.
// ---------------------------------------------------------------------------
__global__ void k_dwconv3x3_silu(const float* __restrict__ uz, const float* __restrict__ cw,
                                 const float* __restrict__ cb, float* __restrict__ u2)
{
    int gid = blockIdx.x * blockDim.x + threadIdx.x;
    if (gid >= BQ * LQ * DIQ) return;
    int d = gid % DIQ;
    int l = (gid / DIQ) % LQ;
    int b = gid / (DIQ * LQ);
    int h = l >> 5, w = l & 31;
    float acc = cb[d];
    #pragma unroll
    for (int ky = 0; ky < 3; ++ky)
        #pragma unroll
        for (int kx = 0; kx < 3; ++kx) {
            int hh = h + ky - 1, ww = w + kx - 1;
            if (hh >= 0 && hh < 32 && ww >= 0 && ww < 32)
                acc += uz[((long)(b * LQ + hh * 32 + ww)) * 512 + d] * cw[d * 9 + ky * 3 + kx];
        }
    u2[gid] = silu_f(acc);
}

// ---------------------------------------------------------------------------
// Build 4-direction sequences us (b,4,L,DI) from u2 (b,L,DI).
// ---------------------------------------------------------------------------
__global__ void k_build_us(const float* __restrict__ u2, float* __restrict__ us)
{
    int gid = blockIdx.x * blockDim.x + threadIdx.x;
    if (gid >= BQ * LQ * DIQ) return;
    int d = gid % DIQ;
    int t = (gid / DIQ) % LQ;
    int b = gid / (DIQ * LQ);
    long ob = (long)b * 4 * LQ * DIQ;
    int l1 = (t & 31) * 32 + (t >> 5);           // transposed (w-major) traversal
    int t3 = LQ - 1 - t;
    int l3 = (t3 & 31) * 32 + (t3 >> 5);
    us[ob + 0L * LQ * DIQ + (long)t * DIQ + d] = u2[((long)(b * LQ + t)) * DIQ + d];
    us[ob + 1L * LQ * DIQ + (long)t * DIQ + d] = u2[((long)(b * LQ + l1)) * DIQ + d];
    us[ob + 2L * LQ * DIQ + (long)t * DIQ + d] = u2[((long)(b * LQ + (LQ - 1 - t))) * DIQ + d];
    us[ob + 3L * LQ * DIQ + (long)t * DIQ + d] = u2[((long)(b * LQ + l3)) * DIQ + d];
}

// ---------------------------------------------------------------------------
// Selective scan. Each 16-lane group owns one channel d (16 s-states).
// dA/dBu computed on the fly; h*C reduced over s via shfl_xor butterfly.
// grid = nb*16 blocks, blockDim 256 (8 waves -> 16 channels per block).
// ---------------------------------------------------------------------------
__global__ void k_scan(const float* __restrict__ dt, const float* __restrict__ u,
                       const float* __restrict__ dbc,
                       const float* __restrict__ Alog, long batchAlog,
                       const float* __restrict__ Dp, long batchDp, int bmod,
                       float* __restrict__ y, int nb)
{
    int wave = threadIdx.x >> 5;
    int lane = threadIdx.x & 31;
    int s    = lane & 15;
    int dsel = lane >> 4;
    int dblk = blockIdx.x % (DIQ / 16);
    int bb   = blockIdx.x / (DIQ / 16);
    if (bb >= nb) return;
    int d    = dblk * 16 + wave * 2 + dsel;
    int aidx = bmod ? (bb % bmod) : 0;

    float a  = -expf(Alog[(long)aidx * batchAlog + d * DSQ + s]);
    float Dv = Dp[(long)aidx * batchDp + d];

    const float* dtp = dt  + (long)bb * LQ * DIQ + d;
    const float* up  = u   + (long)bb * LQ * DIQ + d;
    const float* dbp = dbc + (long)bb * LQ * 48;
    float*       yp  = y   + (long)bb * LQ * DIQ + d;

    float h = 0.f;
    for (int l = 0; l < LQ; ++l) {
        float dtv = dtp[(long)l * DIQ];
        float uv  = up[(long)l * DIQ];
        float Bv  = dbp[l * 48 + DTRQ + s];
        float Cv  = dbp[l * 48 + DTRQ + DSQ + s];
        float dA  = expf(dtv * a);
        h = dA * h + (dtv * uv) * Bv;
        float p = h * Cv;
        p += __shfl_xor(p, 1, 16);
        p += __shfl_xor(p, 2, 16);
        p += __shfl_xor(p, 4, 16);
        p += __shfl_xor(p, 8, 16);
        if (s == 0) yp[(long)l * DIQ] = p + uv * Dv;
    }
}

// ---------------------------------------------------------------------------
// Mamba gate: g = y * silu(z), z = uz[..., 256:512]
// ---------------------------------------------------------------------------
__global__ void k_gate(const float* __restrict__ y, const float* __restrict__ uz,
                       float* __restrict__ g)
{
    int gid = blockIdx.x * blockDim.x + threadIdx.x;
    if (gid >= BQ * LQ * DIQ) return;
    int d  = gid % DIQ;
    int bl = gid / DIQ;
    float z = uz[(long)bl * 512 + DIQ + d];
    g[gid] = y[gid] * silu_f(z);
}

// ---------------------------------------------------------------------------
// VSS: merge 4 directions back to spatial order, LayerNorm(onorm), gate by z.
// blockDim = 256 (= DI), one (b,l) per block.
// ---------------------------------------------------------------------------
__global__ void k_combine_ln_gate(const float* __restrict__ y, const float* __restrict__ uz,
                                  const float* __restrict__ onw, const float* __restrict__ onb,
                                  float* __restrict__ g)
{
    extern __shared__ float sm[];
    int bl = blockIdx.x;
    int b  = bl / LQ;
    int l  = bl % LQ;
    int d  = threadIdx.x;
    int h  = l >> 5, w = l & 31;
    int t1 = w * 32 + h;
    long base = (long)b * 4 * LQ * DIQ;
    float v = y[base + (long)l * DIQ + d]
            + y[base + 1L * LQ * DIQ + (long)t1 * DIQ + d]
            + y[base + 2L * LQ * DIQ + (long)(LQ - 1 - l) * DIQ + d]
            + y[base + 3L * LQ * DIQ + (long)(LQ - 1 - t1) * DIQ + d];
    sm[d] = v; __syncthreads();
    for (int s = DIQ >> 1; s > 0; s >>= 1) { if (d < s) sm[d] += sm[d + s]; __syncthreads(); }
    float mean = sm[0] / DIQ; __syncthreads();
    float dv = v - mean;
    sm[d] = dv * dv; __syncthreads();
    for (int s = DIQ >> 1; s > 0; s >>= 1) { if (d < s) sm[d] += sm[d + s]; __syncthreads(); }
    float var = sm[0] / DIQ;
    float ym = dv * rsqrtf(var + 1e-5f) * onw[d] + onb[d];
    float z  = uz[(long)bl * 512 + DIQ + d];
    g[(long)bl * DIQ + d] = ym * silu_f(z);
}

// ---------------------------------------------------------------------------
// Fuse 3x3 conv (256->256, pad 1) as implicit GEMM on WMMA + BN + ReLU.
// Loop order: 9 taps outer (boundary logic + mask hoisted), channels inner,
// split into p1-half [0,128) and vx-half [128,256) so every inner-loop load
// is unconditional and linear in c.  Branch-free OOB via clamp + 0/1 mask.
// ---------------------------------------------------------------------------
__global__ void k_fuse(const float* __restrict__ p1, const float* __restrict__ vx,
                       const float* __restrict__ fw, const float* __restrict__ fb,
                       const float* __restrict__ bng, const float* __restrict__ bnb,
                       const float* __restrict__ bnm, const float* __restrict__ bnv,
                       float* __restrict__ out)
{
    int wave = (int)(((long)blockIdx.x * blockDim.x + threadIdx.x) >> 5);
    if (wave >= 128 * 16) return;
    int nt = wave % 16;
    int mt = wave / 16;
    int lane = threadIdx.x & 31;
    int hl   = lane >> 4;
    int idx  = lane & 15;

    int m = mt * 16 + idx;          // A row this lane loads
    int b = m >> 10;
    int l = m & 1023;
    int h = l >> 5, w = l & 31;
    int ncol = nt * 16 + idx;       // B/D column this lane owns

    const float* fwbase = fw + (long)ncol * 2304;

    v8f c = {0.f,0.f,0.f,0.f,0.f,0.f,0.f,0.f};
    for (int p = 0; p < 9; ++p) {
        int ky = p / 3, kx = p - ky * 3;
        int hh = h + ky - 1, ww = w + kx - 1;
        bool ok  = (hh >= 0 && hh < 32 && ww >= 0 && ww < 32);
        int  hhc = hh < 0 ? 0 : (hh > 31 ? 31 : hh);
        int  wwc = ww < 0 ? 0 : (ww > 31 ? 31 : ww);
        float msk = ok ? 1.0f : 0.0f;
        long  so  = ((long)(b * LQ + hhc * 32 + wwc)) * HALFQ;
        const float* s1  = p1 + so + 2 * hl;           // channels [0,128)
        const float* s2  = vx + so + 2 * hl;           // channels [128,256)
        const float* fwp = fwbase + p + (long)(2 * hl) * 9;

        #pragma unroll 4
        for (int c0 = 0; c0 < HALFQ; c0 += 4) {
            v2f a, bv;
            a.x = s1[c0] * msk;
            a.y = s1[c0 + 1] * msk;
            bv.x = fwp[(long)c0 * 9];
            bv.y = fwp[(long)(c0 + 1) * 9];
            c = __builtin_amdgcn_wmma_f32_16x16x4_f32(false, a, false, bv, (short)0, c, false, false);
        }
        const float* fwp2 = fwp + (long)HALFQ * 9;
        #pragma unroll 4
        for (int c0 = 0; c0 < HALFQ; c0 += 4) {
            v2f a, bv;
            a.x = s2[c0] * msk;
            a.y = s2[c0 + 1] * msk;
            bv.x = fwp2[(long)c0 * 9];
            bv.y = fwp2[(long)(c0 + 1) * 9];
            c = __builtin_amdgcn_wmma_f32_16x16x4_f32(false, a, false, bv, (short)0, c, false, false);
        }
    }

    float scale = rsqrtf(bnv[ncol] + 1e-5f) * bng[ncol];
    float fbv = fb[ncol], bmv = bnm[ncol], bbv = bnb[ncol];
    #pragma unroll
    for (int r = 0; r < 8; ++r) {
        int mm = mt * 16 + hl * 8 + r;
        int bo = mm >> 10;
        int lo = mm & 1023;
        float v = (c[r] + fbv - bmv) * scale + bbv;
        v = v > 0.f ? v : 0.f;
        out[(long)bo * DIMQ * LQ + (long)ncol * LQ + lo] = v;
    }
}

// ---------------------------------------------------------------------------
extern "C" void kernel_launch(void* const* d_in, const int* in_sizes, int n_in,
                              void* d_out, int out_size, void* d_ws, size_t ws_size,
                              hipStream_t stream)
{
    (void)in_sizes; (void)n_in; (void)out_size; (void)ws_size;
    const float* x         = (const float*)d_in[0];
    const float* proj1_w   = (const float*)d_in[1];
    const float* proj1_b   = (const float*)d_in[2];
    const float* proj2_w   = (const float*)d_in[3];
    const float* proj2_b   = (const float*)d_in[4];
    const float* m_ln_w    = (const float*)d_in[5];
    const float* m_ln_b    = (const float*)d_in[6];
    const float* m_in_w    = (const float*)d_in[7];
    const float* m_conv_w  = (const float*)d_in[8];
    const float* m_conv_b  = (const float*)d_in[9];
    const float* m_xproj_w = (const float*)d_in[10];
    const float* m_dt_w    = (const float*)d_in[11];
    const float* m_dt_b    = (const float*)d_in[12];
    const float* m_Alog    = (const float*)d_in[13];
    const float* m_D       = (const float*)d_in[14];
    const float* m_out_w   = (const float*)d_in[15];
    const float* v_ln_w    = (const float*)d_in[16];
    const float* v_ln_b    = (const float*)d_in[17];
    const float* v_in_w    = (const float*)d_in[18];
    const float* v_conv_w  = (const float*)d_in[19];
    const float* v_conv_b  = (const float*)d_in[20];
    const float* v_xproj_w = (const float*)d_in[21];
    const float* v_dt_w    = (const float*)d_in[22];
    const float* v_dt_b    = (const float*)d_in[23];
    const float* v_Alog    = (const float*)d_in[24];
    const float* v_D       = (const float*)d_in[25];
    const float* v_onorm_w = (const float*)d_in[26];
    const float* v_onorm_b = (const float*)d_in[27];
    const float* v_out_w   = (const float*)d_in[28];
    const float* fuse_w    = (const float*)d_in[29];
    const float* fuse_b    = (const float*)d_in[30];
    const float* bn_g      = (const float*)d_in[31];
    const float* bn_b      = (const float*)d_in[32];
    const float* bn_m      = (const float*)d_in[33];
    const float* bn_v      = (const float*)d_in[34];

    // ---- workspace layout (floats) ----
    float* ws  = (float*)d_ws;
    float* p1  = ws;                                 // (B,L,128)  mamba-path state
    float* vx  = p1  + (long)BQ * LQ * HALFQ;        // (B,L,128)  vss-path state
    float* hb  = vx  + (long)BQ * LQ * HALFQ;        // (B*L,128)  LN output
    float* uz  = hb  + (long)BQ * LQ * HALFQ;        // (B*L,512)  in_proj output
    float* ub  = uz  + (long)BQ * LQ * 512;          // (B*L,256)  conv+silu output
    float* usb = ub  + (long)BQ * LQ * DIQ;          // (B,4,L,256) vss directions
    float* dbc = usb + (long)BQ * 4 * LQ * DIQ;      // (B[*4],L,48) x_proj output
    float* dtb = dbc + (long)BQ * 4 * LQ * 48;       // (B[*4],L,256) dt
    float* yb  = dtb + (long)BQ * 4 * LQ * DIQ;      // (B[*4],L,256) scan output
    float* gb  = yb  + (long)BQ * 4 * LQ * DIQ;      // (B*L,256)  gated

    auto blk = [](long waves) { return (int)((waves * 32 + 255) / 256); };

    // ---- input projections: (B,L,128) = x^T @ W^T + b ----
    k_gemm<1, LQ, 1, DIMQ, DIMQ, HALFQ, HALFQ, 0, 0><<<blk((long)BQ * 64 * 8), 256, 0, stream>>>(
        x, (long)DIMQ * LQ,
        proj1_w, 0, 0,
        proj1_b, 0,
        p1, (long)LQ * HALFQ,
        BQ, 64, 8);
    k_gemm<1, LQ, 1, DIMQ, DIMQ, HALFQ, HALFQ, 0, 0><<<blk((long)BQ * 64 * 8), 256, 0, stream>>>(
        x, (long)DIMQ * LQ,
        proj2_w, 0, 0,
        proj2_b, 0,
        vx, (long)LQ * HALFQ,
        BQ, 64, 8);

    // ---- Mamba path (DEPTH=2) ----
    for (int i = 0; i < 2; ++i) {
        k_ln<<<BQ * LQ, HALFQ, HALFQ * 4, stream>>>(p1, m_ln_w + i * HALFQ, m_ln_b + i * HALFQ, hb, HALFQ);
        // in_proj: (2048,512) = h(2048,128) @ W(128,512)
        k_gemm<HALFQ, 1, 512, 1, HALFQ, 512, 512, 0, 0><<<blk(128L * 32), 256, 0, stream>>>(
            hb, 0,
            m_in_w + (long)i * HALFQ * 512, 0, 0,
            nullptr, 0,
            uz, 0,
            1, 128, 32);
        k_conv1d_silu<<<(BQ * LQ * DIQ) / 256, 256, 0, stream>>>(uz, m_conv_w + i * DIQ * K1Q, m_conv_b + i * DIQ, ub);
        // x_proj: (2048,40) = u(2048,256) @ W(256,40), row stride padded to 48
        k_gemm<DIQ, 1, 40, 1, DIQ, 48, 40, 0, 0><<<blk(128L * 3), 256, 0, stream>>>(
            ub, 0,
            m_xproj_w + (long)i * DIQ * 40, 0, 0,
            nullptr, 0,
            dbc, 0,
            1, 128, 3);
        // dt = softplus(dtr(2048,8) @ W(8,256) + b)
        k_gemm<48, 1, DIQ, 1, DTRQ, DIQ, DIQ, 0, 1><<<blk(128L * 16), 256, 0, stream>>>(
            dbc, 0,
            m_dt_w + (long)i * DTRQ * DIQ, 0, 0,
            m_dt_b + (long)i * DIQ, 0,
            dtb, 0,
            1, 128, 16);
        k_scan<<<2 * 16, 256, 0, stream>>>(dtb, ub, dbc,
            m_Alog + (long)i * DIQ * DSQ, 0,
            m_D + (long)i * DIQ, 0, 1,
            yb, 2);
        k_gate<<<(BQ * LQ * DIQ) / 256, 256, 0, stream>>>(yb, uz, gb);
        // out_proj, residual accumulate into p1
        k_gemm<DIQ, 1, HALFQ, 1, DIQ, HALFQ, HALFQ, 1, 0><<<blk(128L * 8), 256, 0, stream>>>(
            gb, 0,
            m_out_w + (long)i * DIQ * HALFQ, 0, 0,
            nullptr, 0,
            p1, 0,
            1, 128, 8);
    }

    // ---- VSS path (DEPTH=2, 4 scan directions) ----
    for (int i = 0; i < 2; ++i) {
        k_ln<<<BQ * LQ, HALFQ, HALFQ * 4, stream>>>(vx, v_ln_w + i * HALFQ, v_ln_b + i * HALFQ, hb, HALFQ);
        k_gemm<HALFQ, 1, 512, 1, HALFQ, 512, 512, 0, 0><<<blk(128L * 32), 256, 0, stream>>>(
            hb, 0,
            v_in_w + (long)i * HALFQ * 512, 0, 0,
            nullptr, 0,
            uz, 0,
            1, 128, 32);
        k_dwconv3x3_silu<<<(BQ * LQ * DIQ) / 256, 256, 0, stream>>>(uz, v_conv_w + i * DIQ * 9, v_conv_b + i * DIQ, ub);
        k_build_us<<<(BQ * LQ * DIQ) / 256, 256, 0, stream>>>(ub, usb);
        // x_proj per (b,k): B matrix indexed by k = bb % 4
        k_gemm<DIQ, 1, 40, 1, DIQ, 48, 40, 0, 0><<<blk(8L * 64 * 3), 256, 0, stream>>>(
            usb, (long)LQ * DIQ,
            v_xproj_w + (long)i * 4 * DIQ * 40, (long)DIQ * 40, 4,
            nullptr, 0,
            dbc, (long)LQ * 48,
            8, 64, 3);
        // dt per (b,k)
        k_gemm<48, 1, DIQ, 1, DTRQ, DIQ, DIQ, 0, 1><<<blk(8L * 64 * 16), 256, 0, stream>>>(
            dbc, (long)LQ * 48,
            v_dt_w + (long)i * 4 * DTRQ * DIQ, (long)DTRQ * DIQ, 4,
            v_dt_b + (long)i * 4 * DIQ, DIQ,
            dtb, (long)LQ * DIQ,
            8, 64, 16);
        k_scan<<<8 * 16, 256, 0, stream>>>(dtb, usb, dbc,
            v_Alog + (long)i * 4 * DIQ * DSQ, (long)DIQ * DSQ,
            v_D + (long)i * 4 * DIQ, DIQ, 4,
            yb, 8);
        k_combine_ln_gate<<<BQ * LQ, DIQ, DIQ * 4, stream>>>(yb, uz, v_onorm_w + i * DIQ, v_onorm_b + i * DIQ, gb);
        k_gemm<DIQ, 1, HALFQ, 1, DIQ, HALFQ, HALFQ, 1, 0><<<blk(128L * 8), 256, 0, stream>>>(
            gb, 0,
            v_out_w + (long)i * DIQ * HALFQ, 0, 0,
            nullptr, 0,
            vx, 0,
            1, 128, 8);
    }

    // ---- fuse conv 3x3 + BN + ReLU (implicit GEMM, 2048x256xK2304) ----
    k_fuse<<<blk(128L * 16), 256, 0, stream>>>(p1, vx, fuse_w, fuse_b,
                                               bn_g, bn_b, bn_m, bn_v, (float*)d_out);
}